// TensorParallelLogitsKLLoss_88201448390906
// MI455X (gfx1250) — compile-verified
//
#include <hip/hip_runtime.h>

// ---------------------------------------------------------------------------
// KL(P||Q) over vocab dim, Megatron [S,B,V] layout, out [B,S].
// One 256-thread block (8 wave32) per row; single streaming pass with online
// softmax merges. Memory-bound: ~1.05 GB @ 23.3 TB/s => ~45us floor.
// This revision: 4 independent accumulator states per thread (one per float4
// lane) to break the serial exp dependency chain and quadruple ILP.
// ---------------------------------------------------------------------------

typedef float v4f __attribute__((ext_vector_type(4)));

#define S_DIM   2048
#define B_DIM   2
#define V_DIM   32000
#define THREADS 256
#define NWAVES  (THREADS / 32)
#define V4      (V_DIM / 4)              // 8000 float4 per row
#define FULL_IT (V4 / THREADS)           // 31 full strided iterations
#define TAIL_V4 (V4 - FULL_IT * THREADS) // 64 float4 tail
#define PF_DIST 8                        // prefetch ~32KB ahead per tensor

// Online update: one v_exp_f32 per element per side.
// d = v - m ; x = exp(-|d|). If d<=0: term=x, scale=1 else term=1, scale=x.
__device__ __forceinline__ void upd_q(float q, float& mq, float& zq) {
    float d  = q - mq;                       // +inf on first element (mq=-inf)
    float mn = fmaxf(mq, q);
    float x  = __expf(-fabsf(d));
    float e  = (d <= 0.f) ? x : 1.f;
    float c  = (d <= 0.f) ? 1.f : x;
    zq = __builtin_fmaf(zq, c, e);
    mq = mn;
}

__device__ __forceinline__ void upd_p(float t, float q,
                                      float& mp, float& zp, float& ap) {
    float d  = t - mp;
    float mn = fmaxf(mp, t);
    float x  = __expf(-fabsf(d));
    float e  = (d <= 0.f) ? x : 1.f;
    float c  = (d <= 0.f) ? 1.f : x;
    zp = __builtin_fmaf(zp, c, e);
    ap = __builtin_fmaf(ap, c, e * (t - q));
    mp = mn;
}

// Merge two online-softmax partial states.
__device__ __forceinline__ void merge3(float& m, float& z, float& a,
                                       float mo, float zo, float ao) {
    float mn = fmaxf(m, mo);
    float s0 = __expf(m - mn);
    float s1 = __expf(mo - mn);
    z = __builtin_fmaf(z, s0, zo * s1);
    a = __builtin_fmaf(a, s0, ao * s1);
    m = mn;
}

__device__ __forceinline__ void merge2(float& m, float& z, float mo, float zo) {
    float mn = fmaxf(m, mo);
    z = __builtin_fmaf(z, __expf(m - mn), zo * __expf(mo - mn));
    m = mn;
}

__global__ __launch_bounds__(THREADS)
void kl_loss_rows(const float* __restrict__ preds,   // Q logits [S,B,V]
                  const float* __restrict__ tgts,    // P logits [S,B,V]
                  float* __restrict__ out)           // [B,S]
{
    const int  row   = blockIdx.x;                   // 0 .. S*B-1
    const int  tid   = threadIdx.x;
    const long long base4 = (long long)row * V4;
    const v4f* __restrict__ Q4 = (const v4f*)preds + base4;
    const v4f* __restrict__ P4 = (const v4f*)tgts  + base4;

    // 4 independent online-softmax states (one per float4 lane) for ILP.
    float mp[4], zp[4], ap[4], mq[4], zq[4];
#pragma unroll
    for (int k = 0; k < 4; ++k) {
        mp[k] = -INFINITY; zp[k] = 0.f; ap[k] = 0.f;
        mq[k] = -INFINITY; zq[k] = 0.f;
    }

    // Software-pipelined streaming loop: 128-bit loads, double-buffered.
    v4f tcur = P4[tid];
    v4f qcur = Q4[tid];
#pragma unroll 1
    for (int i = 0; i < FULL_IT; ++i) {
        if (i + PF_DIST < FULL_IT) {
            __builtin_prefetch(&P4[(i + PF_DIST) * THREADS + tid], 0, 3);
            __builtin_prefetch(&Q4[(i + PF_DIST) * THREADS + tid], 0, 3);
        }
        v4f tn = tcur, qn = qcur;
        if (i + 1 < FULL_IT) {
            tn = P4[(i + 1) * THREADS + tid];
            qn = Q4[(i + 1) * THREADS + tid];
        }
#pragma unroll
        for (int k = 0; k < 4; ++k) {
            upd_q(qcur[k], mq[k], zq[k]);
            upd_p(tcur[k], qcur[k], mp[k], zp[k], ap[k]);
        }
        tcur = tn;
        qcur = qn;
    }

    // Tail: 64 remaining float4 (threads 0..63, one b128 load each).
    if (tid < TAIL_V4) {
        v4f tt = P4[FULL_IT * THREADS + tid];
        v4f qt = Q4[FULL_IT * THREADS + tid];
#pragma unroll
        for (int k = 0; k < 4; ++k) {
            upd_q(qt[k], mq[k], zq[k]);
            upd_p(tt[k], qt[k], mp[k], zp[k], ap[k]);
        }
    }

    // Fold the 4 per-thread states into one.
    float Mp = mp[0], Zp = zp[0], Ap = ap[0];
    float Mq = mq[0], Zq = zq[0];
#pragma unroll
    for (int k = 1; k < 4; ++k) {
        merge3(Mp, Zp, Ap, mp[k], zp[k], ap[k]);
        merge2(Mq, Zq, mq[k], zq[k]);
    }

    // Intra-wave (wave32) butterfly reduction.
#pragma unroll
    for (int off = 16; off > 0; off >>= 1) {
        float mo = __shfl_xor(Mp, off, 32);
        float zo = __shfl_xor(Zp, off, 32);
        float ao = __shfl_xor(Ap, off, 32);
        merge3(Mp, Zp, Ap, mo, zo, ao);
        float mo2 = __shfl_xor(Mq, off, 32);
        float zo2 = __shfl_xor(Zq, off, 32);
        merge2(Mq, Zq, mo2, zo2);
    }

    // Cross-wave reduction through LDS (8 partial states).
    __shared__ float red[NWAVES][5];
    const int wave = tid >> 5;
    const int lane = tid & 31;
    if (lane == 0) {
        red[wave][0] = Mp; red[wave][1] = Zp; red[wave][2] = Ap;
        red[wave][3] = Mq; red[wave][4] = Zq;
    }
    __syncthreads();

    if (tid == 0) {
        float mP = red[0][0], zP = red[0][1], aP = red[0][2];
        float mQ = red[0][3], zQ = red[0][4];
#pragma unroll
        for (int w = 1; w < NWAVES; ++w) {
            merge3(mP, zP, aP, red[w][0], red[w][1], red[w][2]);
            merge2(mQ, zQ, red[w][3], red[w][4]);
        }
        // loss = A/Zp - mp - ln Zp + mq + ln Zq
        float loss = aP / zP - mP - __logf(zP) + mQ + __logf(zQ);
        const int s = row / B_DIM;
        const int b = row % B_DIM;
        out[(long long)b * S_DIM + s] = loss;
    }
}

extern "C" void kernel_launch(void* const* d_in, const int* in_sizes, int n_in,
                              void* d_out, int out_size, void* d_ws, size_t ws_size,
                              hipStream_t stream) {
    (void)in_sizes; (void)n_in; (void)out_size; (void)d_ws; (void)ws_size;
    const float* preds = (const float*)d_in[0];   // predictions (student Q)
    const float* tgts  = (const float*)d_in[1];   // targets (teacher P)
    float* out = (float*)d_out;                   // [B, S] float32

    kl_loss_rows<<<dim3(S_DIM * B_DIM), dim3(THREADS), 0, stream>>>(preds, tgts, out);
}